// SwitchingLinear_13915694039649
// MI455X (gfx1250) — compile-verified
//
#include <hip/hip_runtime.h>

typedef float v2f __attribute__((ext_vector_type(2)));
typedef float v8f __attribute__((ext_vector_type(8)));

#define BATCH 2048
#define NEXP  64
#define INF_  512
#define OUTF  512

// ws layout (int32):
//   [0,   64)  counts
//   [64, 129)  offsets (exclusive prefix + total)
//   [129,193)  cursors
//   [256,2304) perm (row ids grouped by expert)

__global__ void sl_zero(int* __restrict__ ws) {
    int i = blockIdx.x * blockDim.x + threadIdx.x;
    if (i < 256) ws[i] = 0;
}

__global__ void sl_hist(const int* __restrict__ idx, int* __restrict__ ws) {
    int b = blockIdx.x * blockDim.x + threadIdx.x;
    if (b < BATCH) atomicAdd(&ws[idx[b]], 1);
}

__global__ void sl_scan(int* __restrict__ ws) {
    if (threadIdx.x == 0) {
        int acc = 0;
        for (int c = 0; c < NEXP; ++c) {
            ws[64 + c]  = acc;   // offsets
            ws[129 + c] = acc;   // cursors
            acc += ws[c];
        }
        ws[64 + NEXP] = acc;
    }
}

__global__ void sl_scatter(const int* __restrict__ idx, int* __restrict__ ws) {
    int b = blockIdx.x * blockDim.x + threadIdx.x;
    if (b < BATCH) {
        int c = idx[b];
        int pos = atomicAdd(&ws[129 + c], 1);
        ws[256 + pos] = b;
    }
}

// grid = (64 experts, OUTF/64 tiles), block = 128 (4 waves).
// Each wave owns a 16-wide OUT slice; loops over 16-row sample tiles of its
// expert group, accumulating D(16x16) = A(16x4) x B(4x16) + D over IN=512.
__global__ void __launch_bounds__(128)
sl_gemm(const float* __restrict__ x, const float* __restrict__ w,
        const int* __restrict__ ws, float* __restrict__ out) {
    const int c  = blockIdx.x;
    const int gs = ws[64 + c];
    const int Nc = ws[64 + c + 1] - gs;
    if (Nc == 0) return;
    const int* __restrict__ perm = ws + 256;

    const unsigned lane = threadIdx.x & 31u;
    const unsigned wave = threadIdx.x >> 5;
    const unsigned nb   = blockIdx.y * 64u + wave * 16u;   // OUT column base

    const unsigned m   = lane & 15u;   // A: row M / B: col N / D: col N
    const unsigned sel = lane >> 4;    // 0 -> K={0,1}, 1 -> K={2,3}
    const unsigned ko  = sel * 2u;

    // B fragment source: W[c][nb+m][*] is contiguous along IN
    const float* __restrict__ wrow = w + ((size_t)(unsigned)c * OUTF + nb + m) * INF_;
    float* __restrict__ outcol = out + nb + m;

    const int nTiles = (Nc + 15) >> 4;
    for (int rt = 0; rt < nTiles; ++rt) {
        const int slot = rt * 16 + (int)m;
        // lane m holds the row id for tile slot m (lanes 16-31 duplicate)
        const int row  = perm[gs + (slot < Nc ? slot : 0)];
        // row in [0, 2048): x offset fits in unsigned 32-bit
        const float* __restrict__ xrow = x + (unsigned)row * (unsigned)INF_;

        v8f acc = {};
        #pragma unroll 8
        for (unsigned k = 0; k < INF_; k += 4) {
            v2f a  = *(const v2f*)(xrow + k + ko);
            v2f bm = *(const v2f*)(wrow + k + ko);
            // (neg_a, A, neg_b, B, c_mod, C, reuse_a, reuse_b)
            acc = __builtin_amdgcn_wmma_f32_16x16x4_f32(
                false, a, false, bm, (short)0, acc, false, false);
        }

        // D layout: VGPR r -> M = r + 8*sel, N = m.
        // Row id for slot M comes from lane M via ds_bpermute (no reload).
        // 32-bit unsigned offsets -> SADDR-base + VGPR-offset stores.
        if (rt * 16 + 16 <= Nc) {
            // full tile: branch-free epilogue
            #pragma unroll
            for (unsigned r = 0; r < 8; ++r) {
                const unsigned M    = r + sel * 8u;
                const unsigned orow = (unsigned)__shfl(row, (int)M, 32);
                outcol[orow * (unsigned)OUTF] = acc[r];
            }
        } else {
            // ragged tail tile
            #pragma unroll
            for (unsigned r = 0; r < 8; ++r) {
                const unsigned M    = r + sel * 8u;
                const unsigned orow = (unsigned)__shfl(row, (int)M, 32);
                if (rt * 16 + (int)M < Nc) {
                    outcol[orow * (unsigned)OUTF] = acc[r];
                }
            }
        }
    }
}

extern "C" void kernel_launch(void* const* d_in, const int* in_sizes, int n_in,
                              void* d_out, int out_size, void* d_ws, size_t ws_size,
                              hipStream_t stream) {
    const float* x   = (const float*)d_in[0];
    const int*   idx = (const int*)d_in[1];
    const float* w   = (const float*)d_in[2];
    float*       out = (float*)d_out;
    int*         ws  = (int*)d_ws;

    sl_zero<<<1, 256, 0, stream>>>(ws);
    sl_hist<<<BATCH / 256, 256, 0, stream>>>(idx, ws);
    sl_scan<<<1, 64, 0, stream>>>(ws);
    sl_scatter<<<BATCH / 256, 256, 0, stream>>>(idx, ws);

    dim3 grid(NEXP, OUTF / 64);
    sl_gemm<<<grid, 128, 0, stream>>>(x, w, ws, out);
}